// HeteroGCNCiteerDropout_40759239639280
// MI455X (gfx1250) — compile-verified
//
#include <hip/hip_runtime.h>
#include <cstdint>
#include <cstddef>

#define NP 100000
#define NA 50000
#define DIN 256
#define DHID 128
#define DOUT 64
#define EC 1000000
#define EW 500000
#define EB 500000

typedef __attribute__((ext_vector_type(2))) float v2f;
typedef __attribute__((ext_vector_type(8))) float v8f;

__device__ __forceinline__ unsigned mix32(unsigned x) {
  x ^= x >> 16; x *= 0x7feb352du;
  x ^= x >> 15; x *= 0x846ca68bu;
  x ^= x >> 16;
  return x;
}

// ---------------- utility kernels ----------------
__global__ void k_zero(float* __restrict__ p, int n) {
  int i = blockIdx.x * blockDim.x + threadIdx.x;
  if (i < n) p[i] = 0.0f;
}
__global__ void k_izero(int* __restrict__ p, int n) {
  int i = blockIdx.x * blockDim.x + threadIdx.x;
  if (i < n) p[i] = 0;
}

// u32 histogram (native global_atomic_add_u32)
__global__ void k_count(const int* __restrict__ key, int E, int* __restrict__ cnt) {
  int i = blockIdx.x * blockDim.x + threadIdx.x;
  if (i < E) atomicAdd(&cnt[key[i]], 1);
}

__global__ void k_norm_from_count(const int* __restrict__ cnt, int n, float* __restrict__ nrm) {
  int i = blockIdx.x * blockDim.x + threadIdx.x;
  if (i < n) nrm[i] = rsqrtf(fmaxf((float)cnt[i], 1.0f));
}

__global__ void k_norm_from_rowptr(const int* __restrict__ rp, int n, float* __restrict__ nrm) {
  int i = blockIdx.x * blockDim.x + threadIdx.x;
  if (i < n) nrm[i] = rsqrtf(fmaxf((float)(rp[i + 1] - rp[i]), 1.0f));
}

// ---------------- two-level exclusive scan (counting sort by dst) ----------------
__global__ __launch_bounds__(256) void k_blocksum(const int* __restrict__ cnt, int n,
                                                  int* __restrict__ bsum) {
  __shared__ int s[256];
  int i = blockIdx.x * 256 + threadIdx.x;
  s[threadIdx.x] = (i < n) ? cnt[i] : 0;
  __syncthreads();
  for (int st = 128; st > 0; st >>= 1) {
    if (threadIdx.x < st) s[threadIdx.x] += s[threadIdx.x + st];
    __syncthreads();
  }
  if (threadIdx.x == 0) bsum[blockIdx.x] = s[0];
}

__global__ __launch_bounds__(1024) void k_scanbsum(int* __restrict__ bsum, int nb) {
  __shared__ int s[1024];
  int v = (threadIdx.x < nb) ? bsum[threadIdx.x] : 0;
  s[threadIdx.x] = v;
  __syncthreads();
  for (int st = 1; st < 1024; st <<= 1) {
    int t = (threadIdx.x >= st) ? s[threadIdx.x - st] : 0;
    __syncthreads();
    s[threadIdx.x] += t;
    __syncthreads();
  }
  if (threadIdx.x < nb) bsum[threadIdx.x] = s[threadIdx.x] - v;  // exclusive
}

// per-block inclusive scan + block offset -> rowptr (exclusive) and cursor copy
__global__ __launch_bounds__(256) void k_scanfinal(const int* __restrict__ cnt, int n,
                                                   const int* __restrict__ bsum,
                                                   int* __restrict__ rowptr,
                                                   int* __restrict__ cursor) {
  __shared__ int s[256];
  int i = blockIdx.x * 256 + threadIdx.x;
  int v = (i < n) ? cnt[i] : 0;
  s[threadIdx.x] = v;
  __syncthreads();
  for (int st = 1; st < 256; st <<= 1) {
    int t = (threadIdx.x >= st) ? s[threadIdx.x - st] : 0;
    __syncthreads();
    s[threadIdx.x] += t;
    __syncthreads();
  }
  if (i < n) {
    int excl = s[threadIdx.x] - v + bsum[blockIdx.x];
    rowptr[i] = excl;
    cursor[i] = excl;
  }
  if (i == n - 1) rowptr[n] = s[threadIdx.x] + bsum[blockIdx.x];  // == E
}

// scatter edge source-ids into CSR slots (1 u32 atomic per edge)
__global__ void k_fill(const int* __restrict__ src, const int* __restrict__ dst, int E,
                       int* __restrict__ cursor, int* __restrict__ eids) {
  int e = blockIdx.x * blockDim.x + threadIdx.x;
  if (e < E) {
    int p = atomicAdd(&cursor[dst[e]], 1);
    eids[p] = src[e];
  }
}

// ---------------- fp32 WMMA GEMM: out[M,NC] = (X * nsrc[row]) @ W ----------------
// block = 128 threads = 4 wave32; each wave computes one 16x16 tile.
#define KC 64
#define ASTR 68   // padded LDS stride: conflict-free fragment reads, rows 16B-aligned

__global__ __launch_bounds__(128)
void k_gemm_norm_wmma(const float* __restrict__ X, const float* __restrict__ nsrc,
                      const float* __restrict__ W, float* __restrict__ out,
                      int K, int NC) {
  __shared__ float As[16 * ASTR];
  const int rowBase = blockIdx.x * 16;
  const int wave = threadIdx.x >> 5;
  const int lane = threadIdx.x & 31;
  const int half = lane >> 4;      // 0: K pair {0,1}, 1: K pair {2,3}
  const int l16  = lane & 15;
  const int col  = blockIdx.y * 64 + wave * 16 + l16;

  v8f acc = {0.f, 0.f, 0.f, 0.f, 0.f, 0.f, 0.f, 0.f};

  for (int k0 = 0; k0 < K; k0 += KC) {
    for (int i = threadIdx.x; i < 16 * (KC / 4); i += 128) {
      const int r  = i >> 4;
      const int c4 = (i & 15) << 2;
      const float4 v = *(const float4*)(X + (size_t)(rowBase + r) * K + k0 + c4);
      const float s = nsrc[rowBase + r];
      float4 sv; sv.x = v.x * s; sv.y = v.y * s; sv.z = v.z * s; sv.w = v.w * s;
      *(float4*)(&As[r * ASTR + c4]) = sv;
    }
    __syncthreads();

    #pragma unroll
    for (int kk = 0; kk < KC; kk += 4) {
      const int ka = kk + (half << 1);
      v2f a; a.x = As[l16 * ASTR + ka];
             a.y = As[l16 * ASTR + ka + 1];
      const int kb = k0 + kk + (half << 1);
      v2f b; b.x = W[(size_t)kb * NC + col];
             b.y = W[(size_t)(kb + 1) * NC + col];
      acc = __builtin_amdgcn_wmma_f32_16x16x4_f32(
          false, a, false, b, (short)0, acc, false, false);
    }
    __syncthreads();
  }

  const int rowOut = rowBase + (half << 3);
  #pragma unroll
  for (int r = 0; r < 8; r++)
    out[(size_t)(rowOut + r) * NC + col] = acc[r];
}

// ---------------- CSR segmented gather-sum: agg[d] += nd[d] * sum_j h[eids[j]] ------
// one wave32 per destination node; lanes stripe features; zero float atomics.
template <int D>
__global__ __launch_bounds__(256)
void k_seg_gather(const float* __restrict__ h, const int* __restrict__ rowptr,
                  const int* __restrict__ eids, const float* __restrict__ ndst,
                  float* __restrict__ agg, int nDst) {
  const int lane = threadIdx.x & 31;
  const int w = (blockIdx.x * blockDim.x + threadIdx.x) >> 5;
  if (w >= nDst) return;
  const int beg = rowptr[w], end = rowptr[w + 1];
  if (beg == end) return;
  constexpr int V = D / 32;
  float acc[V];
  #pragma unroll
  for (int i = 0; i < V; ++i) acc[i] = 0.0f;
  for (int j = beg; j < end; ++j) {
    const int s = eids[j];
    if (j + 1 < end)  // prefetch next gather row (global_prefetch_b8)
      __builtin_prefetch(h + (size_t)eids[j + 1] * D, 0, 1);
    const float* hp = h + (size_t)s * D + lane * V;
    #pragma unroll
    for (int i = 0; i < V; ++i) acc[i] += hp[i];
  }
  const float nw = ndst[w];
  float* ap = agg + (size_t)w * D + lane * V;
  #pragma unroll
  for (int i = 0; i < V; ++i) ap[i] += nw * acc[i];  // exclusive owner: plain RMW
}

// ---------------- epilogues ----------------
__global__ void k_combine_relu_drop(const float* __restrict__ agg,
                                    const float* __restrict__ b0,
                                    const float* __restrict__ b1,
                                    float* __restrict__ out,
                                    int n, int D, unsigned seed) {
  int i = blockIdx.x * blockDim.x + threadIdx.x;
  if (i >= n) return;
  const int c = i % D;
  float v = agg[i] + b0[c] + (b1 ? b1[c] : 0.0f);
  v = fmaxf(v, 0.0f);
  const unsigned r = mix32((unsigned)i ^ seed);
  out[i] = (r & 0x80000000u) ? (v * 2.0f) : 0.0f;   // p_drop = 0.5, scale 1/(1-p)
}

__global__ void k_combine_bias(const float* __restrict__ agg,
                               const float* __restrict__ b0,
                               const float* __restrict__ b1,
                               float* __restrict__ out, int n, int D) {
  int i = blockIdx.x * blockDim.x + threadIdx.x;
  if (i >= n) return;
  const int c = i % D;
  out[i] = agg[i] + b0[c] + (b1 ? b1[c] : 0.0f);
}

// ---------------- driver ----------------
static inline int nblk(long long n, int t) { return (int)((n + t - 1) / t); }

extern "C" void kernel_launch(void* const* d_in, const int* in_sizes, int n_in,
                              void* d_out, int out_size, void* d_ws, size_t ws_size,
                              hipStream_t stream) {
  (void)in_sizes; (void)n_in; (void)out_size; (void)ws_size;
  const float* x_paper   = (const float*)d_in[0];
  const float* x_author  = (const float*)d_in[1];
  const float* W1_cites  = (const float*)d_in[2];
  const float* b1_cites  = (const float*)d_in[3];
  const float* W1_writes = (const float*)d_in[4];
  const float* b1_writes = (const float*)d_in[5];
  const float* W1_wb     = (const float*)d_in[6];
  const float* b1_wb     = (const float*)d_in[7];
  const float* W2_cites  = (const float*)d_in[8];
  const float* b2_cites  = (const float*)d_in[9];
  const float* W2_writes = (const float*)d_in[10];
  const float* b2_writes = (const float*)d_in[11];
  const float* W2_wb     = (const float*)d_in[12];
  const float* b2_wb     = (const float*)d_in[13];
  const int* cites_src   = (const int*)d_in[14];
  const int* cites_dst   = (const int*)d_in[15];
  const int* writes_src  = (const int*)d_in[16];
  const int* writes_dst  = (const int*)d_in[17];
  const int* wb_src      = (const int*)d_in[18];
  const int* wb_dst      = (const int*)d_in[19];

  // ---- float workspace ----
  float* ws = (float*)d_ws;
  size_t off = 0;
  auto carve = [&](size_t n) { float* p = ws + off; off += n; return p; };
  float* ns_c  = carve(NP);   // src (out-deg) norms: 3 contiguous arrays
  float* ns_w  = carve(NA);
  float* ns_wb = carve(NP);
  float* nd_c  = carve(NP);   // dst (in-deg) norms
  float* nd_w  = carve(NP);
  float* nd_wb = carve(NA);
  float* proj  = carve((size_t)NP * DHID);
  float* aggP  = carve((size_t)NP * DHID);
  float* aggA  = carve((size_t)NA * DHID);
  float* hP    = carve((size_t)NP * DHID);
  float* hA    = carve((size_t)NA * DHID);
  float* agg2P = aggP;                           // reuse after combine1
  float* agg2A = aggP + (size_t)NP * DOUT;

  // ---- int workspace (CSR per relation, built once, reused by both layers) ----
  int* ip = (int*)(ws + off);
  size_t ioff = 0;
  auto icarve = [&](size_t n) { int* p = ip + ioff; ioff += n; return p; };
  int* rp_c  = icarve(NP + 1); int* cur_c = icarve(NP); int* eid_c = icarve(EC);
  int* rp_w  = icarve(NP + 1); int* cur_w = icarve(NP); int* eid_w = icarve(EW);
  int* rp_b  = icarve(NA + 1); int* cur_b = icarve(NA); int* eid_b = icarve(EB);
  int* tmpc  = icarve(NP);     // src-degree histogram scratch
  int* bsum  = icarve(1024);

  // ---- build CSR (counting sort by dst) + dst norms ----
  auto build_csr = [&](const int* src, const int* dst, int E, int nDst,
                       int* rp, int* cur, int* eid, float* nd) {
    k_izero<<<nblk(nDst, 256), 256, 0, stream>>>(cur, nDst);
    k_count<<<nblk(E, 256), 256, 0, stream>>>(dst, E, cur);
    const int nb = (nDst + 255) / 256;
    k_blocksum<<<nb, 256, 0, stream>>>(cur, nDst, bsum);
    k_scanbsum<<<1, 1024, 0, stream>>>(bsum, nb);
    k_scanfinal<<<nb, 256, 0, stream>>>(cur, nDst, bsum, rp, cur);
    k_fill<<<nblk(E, 256), 256, 0, stream>>>(src, dst, E, cur, eid);
    k_norm_from_rowptr<<<nblk(nDst, 256), 256, 0, stream>>>(rp, nDst, nd);
  };
  build_csr(cites_src,  cites_dst,  EC, NP, rp_c, cur_c, eid_c, nd_c);
  build_csr(writes_src, writes_dst, EW, NP, rp_w, cur_w, eid_w, nd_w);
  build_csr(wb_src,     wb_dst,     EB, NA, rp_b, cur_b, eid_b, nd_wb);

  // ---- src (out-degree) norms via u32 histograms ----
  k_izero<<<nblk(NP, 256), 256, 0, stream>>>(tmpc, NP);
  k_count<<<nblk(EC, 256), 256, 0, stream>>>(cites_src, EC, tmpc);
  k_norm_from_count<<<nblk(NP, 256), 256, 0, stream>>>(tmpc, NP, ns_c);
  k_izero<<<nblk(NA, 256), 256, 0, stream>>>(tmpc, NA);
  k_count<<<nblk(EW, 256), 256, 0, stream>>>(writes_src, EW, tmpc);
  k_norm_from_count<<<nblk(NA, 256), 256, 0, stream>>>(tmpc, NA, ns_w);
  k_izero<<<nblk(NP, 256), 256, 0, stream>>>(tmpc, NP);
  k_count<<<nblk(EB, 256), 256, 0, stream>>>(wb_src, EB, tmpc);
  k_norm_from_count<<<nblk(NP, 256), 256, 0, stream>>>(tmpc, NP, ns_wb);

  // ---- layer 1 ----
  const long long AGG1 = (long long)NP * DHID + (long long)NA * DHID;
  k_zero<<<nblk(AGG1, 256), 256, 0, stream>>>(aggP, (int)AGG1);

  dim3 g1p(NP / 16, DHID / 64), g1a(NA / 16, DHID / 64);
  k_gemm_norm_wmma<<<g1p, 128, 0, stream>>>(x_paper, ns_c, W1_cites, proj, DIN, DHID);
  k_seg_gather<DHID><<<nblk(NP, 8), 256, 0, stream>>>(proj, rp_c, eid_c, nd_c, aggP, NP);
  k_gemm_norm_wmma<<<g1a, 128, 0, stream>>>(x_author, ns_w, W1_writes, proj, DIN, DHID);
  k_seg_gather<DHID><<<nblk(NP, 8), 256, 0, stream>>>(proj, rp_w, eid_w, nd_w, aggP, NP);
  k_gemm_norm_wmma<<<g1p, 128, 0, stream>>>(x_paper, ns_wb, W1_wb, proj, DIN, DHID);
  k_seg_gather<DHID><<<nblk(NA, 8), 256, 0, stream>>>(proj, rp_b, eid_b, nd_wb, aggA, NA);

  k_combine_relu_drop<<<nblk((long long)NP * DHID, 256), 256, 0, stream>>>(
      aggP, b1_cites, b1_writes, hP, NP * DHID, DHID, 0x2A6A2A6Au);
  k_combine_relu_drop<<<nblk((long long)NA * DHID, 256), 256, 0, stream>>>(
      aggA, b1_wb, nullptr, hA, NA * DHID, DHID, 0x51ED270Bu);

  // ---- layer 2 (reuses the CSRs and norms) ----
  const long long AGG2 = (long long)NP * DOUT + (long long)NA * DOUT;
  k_zero<<<nblk(AGG2, 256), 256, 0, stream>>>(agg2P, (int)AGG2);

  dim3 g2p(NP / 16, DOUT / 64), g2a(NA / 16, DOUT / 64);
  k_gemm_norm_wmma<<<g2p, 128, 0, stream>>>(hP, ns_c, W2_cites, proj, DHID, DOUT);
  k_seg_gather<DOUT><<<nblk(NP, 8), 256, 0, stream>>>(proj, rp_c, eid_c, nd_c, agg2P, NP);
  k_gemm_norm_wmma<<<g2a, 128, 0, stream>>>(hA, ns_w, W2_writes, proj, DHID, DOUT);
  k_seg_gather<DOUT><<<nblk(NP, 8), 256, 0, stream>>>(proj, rp_w, eid_w, nd_w, agg2P, NP);
  k_gemm_norm_wmma<<<g2p, 128, 0, stream>>>(hP, ns_wb, W2_wb, proj, DHID, DOUT);
  k_seg_gather<DOUT><<<nblk(NA, 8), 256, 0, stream>>>(proj, rp_b, eid_b, nd_wb, agg2A, NA);

  float* outP = (float*)d_out;
  float* outA = outP + (size_t)NP * DOUT;
  k_combine_bias<<<nblk((long long)NP * DOUT, 256), 256, 0, stream>>>(
      agg2P, b2_cites, b2_writes, outP, NP * DOUT, DOUT);
  k_combine_bias<<<nblk((long long)NA * DOUT, 256), 256, 0, stream>>>(
      agg2A, b2_wb, nullptr, outA, NA * DOUT, DOUT);
}